// DprnnBlock_skip_22574348107974
// MI455X (gfx1250) — compile-verified
//
#include <hip/hip_runtime.h>
#include <math.h>

// ---------------------------------------------------------------------------
// CDNA5 (gfx1250) fp32 WMMA DPRNN block, round 3.
//  - B matrices pair-interleaved [(k/2)][col][k&1]: each 4x16 B fragment is one
//    aligned b64 load (ds_load_2addr_b64 pairs in the hot loop, no v_mov packing).
//  - xw gate buffers in native wave-fragment layout [tile16][jt][lane][8]:
//    scans reload each 16x16 tile as 2 x b128 per lane.
//  - r3: scale term vectorized (b uniform per 16-tile), fc_ln transpose index
//    hoisted (no per-element /200).
// Fragment conventions (CDNA5 ISA 7.12.2):
//   A 16x4 : lane l holds rows l&15, K = 4kc + 2*(l>>4) + {0,1}
//   B 4x16 : lane l holds col l&15,  K = 4kc + 2*(l>>4) + {0,1}
//   C/D    : vgpr i, lane l -> (M = i + 8*(l>>4), N = l&15)
// ---------------------------------------------------------------------------

typedef float v2f __attribute__((ext_vector_type(2)));
typedef float v8f __attribute__((ext_vector_type(8)));

#define WMMA_F32(A_, B_, C_) \
  __builtin_amdgcn_wmma_f32_16x16x4_f32(false, (A_), false, (B_), (short)0, (C_), false, false)

__device__ __forceinline__ float sigmoidf_(float x) { return 1.0f / (1.0f + __expf(-x)); }

#define XW_INTRA  39321600        // 3200*64*192 floats (one direction)
#define WS_RNN    78643200        // xw region size (== 1024*200*384)
#define WS_IOUT   104857600       // intra_out offset
#define WS_UPK    131072000       // repacked Ui offset (49152 floats)
#define OUT_MAIN  26214400        // B*L*W*C
#define OUT_G1    12800           // 64*L ones
// total ws: 131,121,152 floats = 500.2 MiB

// ---------------------------------------------------------------------------
// Repack a [K,N] row-major matrix into pair-interleaved [(k/2)][n][k&1]
// so B fragments become single b64 loads.
// ---------------------------------------------------------------------------
__global__ void repack_pairs_kernel(const float* __restrict__ src,
                                    float* __restrict__ dst, int K, int N)
{
  int idx = blockIdx.x * blockDim.x + threadIdx.x;
  if (idx < K * N) {
    int k = idx / N, c = idx - k * N;
    dst[((k >> 1) * N + c) * 2 + (k & 1)] = src[idx];
  }
}

// ---------------------------------------------------------------------------
// Kernel 1: xw = rowmap(A)[M,128] @ W[128,N] + bias (+ scale*W[128,:], mode 1)
// M enumerated time-major: r' = t*Nseq + n  (16 consecutive n = one scan tile).
// Output in wave-fragment layout: out[(r'/16)*(N/16) + jtg][lane][8].
// grid = (M/64, N/96), block = 128 (4 waves x 16 rows).
// ---------------------------------------------------------------------------
__global__ __launch_bounds__(128) void gemm_xw_kernel(
    const float* __restrict__ A, const float* __restrict__ Wm,
    const float* __restrict__ bias, const float* __restrict__ scale,
    float* __restrict__ out, int N, int T, int Nseq, int mode)
{
  __shared__ __align__(16) float wlds[128 * 96];   // pair-interleaved (64 pairs x 96 x 2)
  const int tid  = threadIdx.x;
  const int wave = tid >> 5;
  const int lane = tid & 31;
  const int l15  = lane & 15;
  const int mh   = (lane >> 4) << 3;
  const int c0   = blockIdx.y * 96;
  const int rowbase = blockIdx.x * 64 + wave * 16;

  for (int idx = tid; idx < 128 * 96; idx += 128) {
    int k = idx / 96, c = idx - k * 96;
    wlds[((k >> 1) * 96 + c) * 2 + (k & 1)] = Wm[k * N + c0 + c];
  }
  __syncthreads();

  const int t  = rowbase / Nseq;           // uniform across the wave tile
  const int nb = rowbase - t * Nseq;       // first sequence of the tile
  const int nlane = nb + l15;              // A fragment row for this lane

  const float* arow;
  if (mode == 0) {
    arow = A + ((size_t)nlane * T + t) * 128;
  } else {
    int b = nlane >> 6, w = nlane & 63;
    arow = A + ((size_t)((b * 200 + t) * 64 + w)) * 128;
  }

  v8f acc[6];
  #pragma unroll
  for (int jt = 0; jt < 6; ++jt) {
    float bv = bias[c0 + jt * 16 + l15];
    #pragma unroll
    for (int i = 0; i < 8; ++i) acc[jt][i] = bv;
  }

  #pragma unroll 4
  for (int kc = 0; kc < 32; ++kc) {
    v2f a = *(const v2f*)(arow + kc * 4 + ((lane >> 4) << 1));
    const int p = kc * 2 + (lane >> 4);
    #pragma unroll
    for (int jt = 0; jt < 6; ++jt) {
      v2f b = *(const v2f*)(&wlds[(p * 96 + jt * 16 + l15) * 2]);
      acc[jt] = WMMA_F32(a, b, acc[jt]);
    }
  }

  if (mode == 1) {
    // + scale column * Wm[128,:].  A 16-row tile never straddles a 64-block
    // (16 | 64), so b is uniform and the 8 scale values per lane are contiguous.
    const int bscale = nb >> 6;
    const v8f sv = *(const v8f*)(scale + (size_t)(bscale * 200 + t) * 64
                                       + (nb & 63) + mh);
    #pragma unroll
    for (int jt = 0; jt < 6; ++jt) {
      float wl = Wm[128 * N + c0 + jt * 16 + l15];
      #pragma unroll
      for (int i = 0; i < 8; ++i) acc[jt][i] += sv[i] * wl;
    }
  }

  // fragment-layout store: 8 contiguous floats per lane per tile (2 x b128)
  float* orow = out + ((size_t)(rowbase >> 4) * (N >> 4) + (c0 >> 4)) * 256 + lane * 8;
  #pragma unroll
  for (int jt = 0; jt < 6; ++jt)
    *(v8f*)(orow + (size_t)jt * 256) = acc[jt];
}

// ---------------------------------------------------------------------------
// Kernel 2: intra bidirectional GRU scan. One wave per 16-sequence tile.
// grid.x = 400 (blocks 0..199 fwd, 200..399 bwd). H=64, gates=192 (12 tiles).
// ---------------------------------------------------------------------------
__global__ __launch_bounds__(32) void intra_scan_kernel(
    const float* __restrict__ xw, const float* __restrict__ Uf,
    const float* __restrict__ Ub, float* __restrict__ rnn_out)
{
  __shared__ __align__(16) float Ulds[64 * 192];   // pair-interleaved, 48 KB
  __shared__ __align__(16) float hbuf[16 * 64];
  __shared__ __align__(16) float rhbuf[16 * 64];

  const int lane = threadIdx.x;
  const int l15  = lane & 15;
  const int kofs = (lane >> 4) << 1;
  const int mh   = (lane >> 4) << 3;
  const int dir  = (blockIdx.x >= 200) ? 1 : 0;
  const int tile = blockIdx.x % 200;
  const int seqbase = tile * 16;
  const float* U   = dir ? Ub : Uf;
  const float* xwd = xw + (size_t)dir * XW_INTRA;

  for (int idx = lane; idx < 64 * 192; idx += 32) {
    int k = idx / 192, c = idx - k * 192;
    Ulds[((k >> 1) * 192 + c) * 2 + (k & 1)] = U[idx];
  }
  for (int idx = lane; idx < 16 * 64; idx += 32) hbuf[idx] = 0.0f;
  __syncthreads();

  for (int s = 0; s < 64; ++s) {
    const int t = dir ? (63 - s) : s;
    const float* xwt = xwd + ((size_t)t * 200 + tile) * (12 * 256);
    if (s < 63) {
      const float* nxt = xwd + ((size_t)(dir ? t - 1 : t + 1) * 200 + tile) * (12 * 256);
      __builtin_prefetch(nxt + lane * 8, 0, 1);
    }

    v8f acc[12];  // gate pre-activations (fragment layout reload, 2 b128 each)
    #pragma unroll
    for (int jt = 0; jt < 12; ++jt)
      acc[jt] = *(const v8f*)(xwt + (size_t)jt * 256 + lane * 8);

    // z,r: += h @ U[:, 0:128]
    #pragma unroll 2
    for (int kc = 0; kc < 16; ++kc) {
      v2f a = *(const v2f*)(&hbuf[l15 * 64 + kc * 4 + kofs]);
      const int p = kc * 2 + (lane >> 4);
      #pragma unroll
      for (int jt = 0; jt < 8; ++jt) {
        v2f b = *(const v2f*)(&Ulds[(p * 192 + jt * 16 + l15) * 2]);
        acc[jt] = WMMA_F32(a, b, acc[jt]);
      }
    }

    v8f z4[4], hD[4];
    #pragma unroll
    for (int jt = 0; jt < 4; ++jt)
      #pragma unroll
      for (int i = 0; i < 8; ++i) {
        z4[jt][i] = sigmoidf_(acc[jt][i]);
        float rg  = sigmoidf_(acc[jt + 4][i]);
        float hv  = hbuf[(i + mh) * 64 + jt * 16 + l15];
        hD[jt][i] = hv;
        rhbuf[(i + mh) * 64 + jt * 16 + l15] = rg * hv;   // reset BEFORE matmul
      }
    __syncthreads();

    // hh: += (r*h) @ U[:, 128:192]
    #pragma unroll 2
    for (int kc = 0; kc < 16; ++kc) {
      v2f a = *(const v2f*)(&rhbuf[l15 * 64 + kc * 4 + kofs]);
      const int p = kc * 2 + (lane >> 4);
      #pragma unroll
      for (int jt = 0; jt < 4; ++jt) {
        v2f b = *(const v2f*)(&Ulds[(p * 192 + 128 + jt * 16 + l15) * 2]);
        acc[8 + jt] = WMMA_F32(a, b, acc[8 + jt]);
      }
    }
    __syncthreads();

    #pragma unroll
    for (int jt = 0; jt < 4; ++jt)
      #pragma unroll
      for (int i = 0; i < 8; ++i) {
        float hh = tanhf(acc[8 + jt][i]);
        float zz = z4[jt][i];
        float hn = zz * hD[jt][i] + (1.0f - zz) * hh;
        int m = i + mh;
        hbuf[m * 64 + jt * 16 + l15] = hn;
        rnn_out[((size_t)(seqbase + m) * 64 + t) * 128 + dir * 64 + jt * 16 + l15] = hn;
      }
    __syncthreads();
  }
}

// ---------------------------------------------------------------------------
// Kernel 3/6: dense = Ain[M,128] @ Wfc[128,128] + b; out = resid + LN(dense)*g+be
// mode 0: identity row map (intra). mode 1: (b,w,l) -> (b,l,w) transposed store.
// ---------------------------------------------------------------------------
__global__ __launch_bounds__(128) void fc_ln_kernel(
    const float* __restrict__ Ain, const float* __restrict__ Wfc,
    const float* __restrict__ bfc, const float* __restrict__ gamma,
    const float* __restrict__ beta, const float* __restrict__ resid,
    float* __restrict__ out, int mode)
{
  __shared__ __align__(16) float wlds[128 * 128];   // pair-interleaved, 64 KB
  const int tid  = threadIdx.x;
  const int wave = tid >> 5;
  const int lane = tid & 31;
  const int l15  = lane & 15;
  const int mh   = (lane >> 4) << 3;
  const int rowbase = blockIdx.x * 64 + wave * 16;

  for (int idx = tid; idx < 128 * 128; idx += 128) {
    int k = idx >> 7, c = idx & 127;
    wlds[((k >> 1) * 128 + c) * 2 + (k & 1)] = Wfc[idx];
  }
  __syncthreads();

  const float* arow = Ain + (size_t)(rowbase + l15) * 128;

  v8f acc[8];
  #pragma unroll
  for (int jt = 0; jt < 8; ++jt) {
    float bv = bfc[jt * 16 + l15];
    #pragma unroll
    for (int i = 0; i < 8; ++i) acc[jt][i] = bv;
  }

  #pragma unroll 4
  for (int kc = 0; kc < 32; ++kc) {
    v2f a = *(const v2f*)(arow + kc * 4 + ((lane >> 4) << 1));
    const int p = kc * 2 + (lane >> 4);
    #pragma unroll
    for (int jt = 0; jt < 8; ++jt) {
      v2f b = *(const v2f*)(&wlds[(p * 128 + jt * 16 + l15) * 2]);
      acc[jt] = WMMA_F32(a, b, acc[jt]);
    }
  }

  // per-row mean/var: rows 0-7 live in lanes 0-15, rows 8-15 in lanes 16-31
  float sum8[8], sq8[8];
  #pragma unroll
  for (int i = 0; i < 8; ++i) { sum8[i] = 0.0f; sq8[i] = 0.0f; }
  #pragma unroll
  for (int jt = 0; jt < 8; ++jt)
    #pragma unroll
    for (int i = 0; i < 8; ++i) { float v = acc[jt][i]; sum8[i] += v; sq8[i] += v * v; }
  #pragma unroll
  for (int off = 1; off <= 8; off <<= 1)
    #pragma unroll
    for (int i = 0; i < 8; ++i) {
      sum8[i] += __shfl_xor(sum8[i], off, 32);
      sq8[i]  += __shfl_xor(sq8[i],  off, 32);
    }
  float mean8[8], inv8[8];
  #pragma unroll
  for (int i = 0; i < 8; ++i) {
    float mu  = sum8[i] * (1.0f / 128.0f);
    float var = sq8[i] * (1.0f / 128.0f) - mu * mu;
    mean8[i] = mu;
    inv8[i]  = rsqrtf(var + 1e-8f);
  }

  // transpose map hoisted: a 16-row tile crosses at most one l=200 boundary
  const int bw0 = rowbase / 200;
  const int l0  = rowbase - bw0 * 200;

  #pragma unroll
  for (int jt = 0; jt < 8; ++jt) {
    int c = jt * 16 + l15;
    float g = gamma[c], be = beta[c];
    #pragma unroll
    for (int i = 0; i < 8; ++i) {
      float val = (acc[jt][i] - mean8[i]) * inv8[i] * g + be;
      int m = i + mh;
      size_t idx;
      if (mode == 0) {
        idx = (size_t)(rowbase + m) * 128 + c;
      } else {
        int l  = l0 + m;
        int bw = bw0 + (l >= 200 ? 1 : 0);
        l -= (l >= 200 ? 200 : 0);
        int b = bw >> 6, w = bw & 63;
        idx = ((size_t)((b * 200 + l) * 64 + w)) * 128 + c;
      }
      out[idx] = resid[idx] + val;
    }
  }
}

// ---------------------------------------------------------------------------
// Kernel 5: inter skip-GRU scan. H=128, gates=384 (24 tiles), T=200.
// 64 single-wave blocks; Ui pre-repacked to pair layout (b64 fragments).
// ---------------------------------------------------------------------------
__global__ __launch_bounds__(32) void inter_scan_kernel(
    const float* __restrict__ xw, const float* __restrict__ Upk,
    const float* __restrict__ Wp, const float* __restrict__ bp,
    float* __restrict__ rnn_out, float* __restrict__ gate_out)
{
  __shared__ __align__(16) float hbuf[16 * 128];
  __shared__ __align__(16) float rhbuf[16 * 128];
  __shared__ float Wplds[128];
  __shared__ float uts[16];
  __shared__ float us[16];

  const int lane = threadIdx.x;
  const int l15  = lane & 15;
  const int kofs = (lane >> 4) << 1;
  const int mh   = (lane >> 4) << 3;
  const int tile = blockIdx.x;
  const int seqbase = tile * 16;

  for (int idx = lane; idx < 16 * 128; idx += 32) hbuf[idx] = 0.0f;
  for (int idx = lane; idx < 128; idx += 32) Wplds[idx] = Wp[idx];
  if (lane < 16) uts[lane] = 1.0f;
  const float bpv = bp[0];
  __syncthreads();

  for (int t = 0; t < 200; ++t) {
    const float* xwt = xw + ((size_t)t * 64 + tile) * (24 * 256);
    if (t < 199) __builtin_prefetch(xwt + (size_t)64 * 24 * 256 + lane * 8, 0, 1);

    if (lane < 16) us[lane] = rintf(uts[lane]);   // u = round(ut), RNE == jnp.round
    __syncthreads();

    v8f acc[16];  // z (tiles 0-7) and r (tiles 8-15)
    #pragma unroll
    for (int jt = 0; jt < 16; ++jt)
      acc[jt] = *(const v8f*)(xwt + (size_t)jt * 256 + lane * 8);

    #pragma unroll 1
    for (int kc = 0; kc < 32; ++kc) {
      v2f a = *(const v2f*)(&hbuf[l15 * 128 + kc * 4 + kofs]);
      const int p = kc * 2 + (lane >> 4);
      #pragma unroll
      for (int jt = 0; jt < 16; ++jt) {
        v2f b = *(const v2f*)(&Upk[(size_t)(p * 384 + jt * 16 + l15) * 2]);
        acc[jt] = WMMA_F32(a, b, acc[jt]);
      }
    }

    #pragma unroll
    for (int jt = 0; jt < 8; ++jt)
      #pragma unroll
      for (int i = 0; i < 8; ++i) {
        acc[jt][i] = sigmoidf_(acc[jt][i]);                 // z in place
        float rg = sigmoidf_(acc[8 + jt][i]);
        rhbuf[(i + mh) * 128 + jt * 16 + l15] =
            rg * hbuf[(i + mh) * 128 + jt * 16 + l15];
      }
    __syncthreads();

    v8f acch[8];
    #pragma unroll
    for (int jt = 0; jt < 8; ++jt)
      acch[jt] = *(const v8f*)(xwt + (size_t)(16 + jt) * 256 + lane * 8);

    #pragma unroll 1
    for (int kc = 0; kc < 32; ++kc) {
      v2f a = *(const v2f*)(&rhbuf[l15 * 128 + kc * 4 + kofs]);
      const int p = kc * 2 + (lane >> 4);
      #pragma unroll
      for (int jt = 0; jt < 8; ++jt) {
        v2f b = *(const v2f*)(&Upk[(size_t)(p * 384 + 256 + jt * 16 + l15) * 2]);
        acch[jt] = WMMA_F32(a, b, acch[jt]);
      }
    }
    __syncthreads();

    #pragma unroll
    for (int jt = 0; jt < 8; ++jt)
      #pragma unroll
      for (int i = 0; i < 8; ++i) {
        int m = i + mh;
        float hold = hbuf[m * 128 + jt * 16 + l15];
        float hh = tanhf(acch[jt][i]);
        float zz = acc[jt][i];
        float hn = zz * hold + (1.0f - zz) * hh;
        float u  = us[m];
        float hf = u * hn + (1.0f - u) * hold;     // skip gate
        hbuf[m * 128 + jt * 16 + l15] = hf;
        rnn_out[((size_t)(seqbase + m) * 200 + t) * 128 + jt * 16 + l15] = hf;
      }
    __syncthreads();

    if (lane < 16) {   // delta = sigmoid(h @ Wp + bp); ut update; emit gate
      float dot = bpv;
      for (int k = 0; k < 128; ++k) dot += hbuf[lane * 128 + k] * Wplds[k];
      float delta = sigmoidf_(dot);
      float u  = us[lane];
      float uo = uts[lane];
      uts[lane] = u * delta + (1.0f - u) * (uo + fminf(delta, 1.0f - uo));
      gate_out[(size_t)(seqbase + lane) * 200 + t] = u;
    }
    __syncthreads();
  }
}

__global__ void fill_ones_kernel(float* __restrict__ p, int n)
{
  int i = blockIdx.x * blockDim.x + threadIdx.x;
  if (i < n) p[i] = 1.0f;
}

// ---------------------------------------------------------------------------
extern "C" void kernel_launch(void* const* d_in, const int* in_sizes, int n_in,
                              void* d_out, int out_size, void* d_ws, size_t ws_size,
                              hipStream_t stream) {
  const float* x      = (const float*)d_in[0];
  const float* scale  = (const float*)d_in[1];
  const float* Wf     = (const float*)d_in[2];
  const float* Uf     = (const float*)d_in[3];
  const float* bf     = (const float*)d_in[4];
  const float* Wb     = (const float*)d_in[5];
  const float* Ub     = (const float*)d_in[6];
  const float* bb     = (const float*)d_in[7];
  const float* W_ifc  = (const float*)d_in[8];
  const float* b_ifc  = (const float*)d_in[9];
  const float* g_i    = (const float*)d_in[10];
  const float* be_i   = (const float*)d_in[11];
  const float* Wi     = (const float*)d_in[12];
  const float* Ui     = (const float*)d_in[13];
  const float* bi     = (const float*)d_in[14];
  const float* Wp     = (const float*)d_in[15];
  const float* bp     = (const float*)d_in[16];
  const float* W_efc  = (const float*)d_in[17];
  const float* b_efc  = (const float*)d_in[18];
  const float* g_e    = (const float*)d_in[19];
  const float* be_e   = (const float*)d_in[20];

  float* out = (float*)d_out;
  float* ws  = (float*)d_ws;
  float* xwbuf     = ws;             // 78,643,200 floats (shared intra/inter)
  float* rnnbuf    = ws + WS_RNN;    // 26,214,400 floats (intra then inter rnn out)
  float* intra_out = ws + WS_IOUT;   // 26,214,400 floats
  float* upk       = ws + WS_UPK;    // 49,152 floats (pair-interleaved Ui)
  float* gate2     = out + OUT_MAIN + OUT_G1;

  dim3 blk(128);

  // repack Ui once per call (input is constant across the call)
  repack_pairs_kernel<<<(128 * 384 + 255) / 256, 256, 0, stream>>>(Ui, upk, 128, 384);

  // intra: xw for both directions (fragment layout), then the bidirectional scan
  gemm_xw_kernel<<<dim3(3200, 2), blk, 0, stream>>>(x, Wf, bf, nullptr, xwbuf,            192, 64, 3200, 0);
  gemm_xw_kernel<<<dim3(3200, 2), blk, 0, stream>>>(x, Wb, bb, nullptr, xwbuf + XW_INTRA, 192, 64, 3200, 0);
  intra_scan_kernel<<<400, 32, 0, stream>>>(xwbuf, Uf, Ub, rnnbuf);
  fc_ln_kernel<<<3200, 128, 0, stream>>>(rnnbuf, W_ifc, b_ifc, g_i, be_i, x, intra_out, 0);

  // inter: transposed read + scale column fused into the xw GEMM, then skip-GRU
  gemm_xw_kernel<<<dim3(3200, 4), blk, 0, stream>>>(intra_out, Wi, bi, scale, xwbuf, 384, 200, 1024, 1);
  inter_scan_kernel<<<64, 32, 0, stream>>>(xwbuf, upk, Wp, bp, rnnbuf, gate2);
  fc_ln_kernel<<<3200, 128, 0, stream>>>(rnnbuf, W_efc, b_efc, g_e, be_e, intra_out, out, 1);

  fill_ones_kernel<<<(OUT_G1 + 255) / 256, 256, 0, stream>>>(out + OUT_MAIN, OUT_G1);
}